// GATLayer_38019050504251
// MI455X (gfx1250) — compile-verified
//
#include <hip/hip_runtime.h>
#include <math.h>

#define BB 16
#define NN 1024
#define IN_DIM 256
#define OUT_DIM 256
#define HH 4
#define DD 64
#define ROWS (BB * NN) /* 16384 */

typedef __attribute__((ext_vector_type(2))) float v2f;
typedef __attribute__((ext_vector_type(8))) float v8f;

__device__ __forceinline__ v8f wmma_f32(v2f a, v2f b, v8f c) {
  // D = A(16x4) * B(4x16) + C(16x16), fp32, wave32
  return __builtin_amdgcn_wmma_f32_16x16x4_f32(false, a, false, b, (short)0, c,
                                               false, false);
}

__device__ __forceinline__ v8f v8zero() {
  v8f z = {0.f, 0.f, 0.f, 0.f, 0.f, 0.f, 0.f, 0.f};
  return z;
}

// ---- CDNA5 async global->LDS DMA (ASYNCcnt-tracked, bypasses VGPRs) ----
__device__ __forceinline__ void async_ld_b128(unsigned lds_addr,
                                              const float* gaddr) {
  asm volatile("global_load_async_to_lds_b128 %0, %1, off"
               :
               : "v"(lds_addr), "v"(gaddr)
               : "memory");
}
__device__ __forceinline__ void wait_async0() {
  asm volatile("s_wait_asynccnt 0" ::: "memory");
}
__device__ __forceinline__ unsigned lds_off(const void* p) {
  return (unsigned)(uintptr_t)p; // low 32 bits of generic ptr == LDS offset
}

// ---------------- Kernel 1: Wh = h @ W  (fp32 WMMA) ----------------
__global__ __launch_bounds__(128) void k_wh(const float* __restrict__ h,
                                            const float* __restrict__ W,
                                            float* __restrict__ Wh) {
  // pitch 260: 16B-aligned rows for b128 DMA, conflict-free column reads
  __shared__ __align__(16) float hs[16][260];
  const int tid = threadIdx.x;
  const int m0 = blockIdx.x * 16;
#pragma unroll
  for (int i = 0; i < 8; ++i) {
    int idx = tid + i * 128; // 1024 x b128 chunks
    int r = idx >> 6, c4 = (idx & 63) * 4;
    async_ld_b128(lds_off(&hs[r][c4]), h + (size_t)(m0 + r) * IN_DIM + c4);
  }
  wait_async0();
  __syncthreads();

  const int lane = tid & 31;
  const int wave = tid >> 5;
  const int half = lane >> 4;
  const int l = lane & 15;
  const int colbase = wave * 64;

  v8f acc[4];
#pragma unroll
  for (int dt = 0; dt < 4; ++dt) acc[dt] = v8zero();

  for (int k0 = 0; k0 < IN_DIM; k0 += 4) {
    const int ka = k0 + 2 * half; // v0: K=k0|k0+2 ; v1: K=k0+1|k0+3
    v2f a;
    a.x = hs[l][ka];
    a.y = hs[l][ka + 1];
#pragma unroll
    for (int dt = 0; dt < 4; ++dt) {
      const float* wp = W + (size_t)ka * OUT_DIM + colbase + dt * 16 + l;
      v2f bf;
      bf.x = wp[0];
      bf.y = wp[OUT_DIM];
      acc[dt] = wmma_f32(a, bf, acc[dt]);
    }
  }
#pragma unroll
  for (int dt = 0; dt < 4; ++dt)
#pragma unroll
    for (int j = 0; j < 8; ++j) {
      int gr = m0 + j + half * 8;
      Wh[(size_t)gr * OUT_DIM + colbase + dt * 16 + l] = acc[dt][j];
    }
}

// ---------------- Kernel 2: e_src / e_dst per (row, head) ----------------
__global__ __launch_bounds__(256) void k_attcoef(
    const float* __restrict__ Wh, const float* __restrict__ a_src,
    const float* __restrict__ a_dst, float* __restrict__ e_src,
    float* __restrict__ e_dst) {
  int t = blockIdx.x * 256 + threadIdx.x; // 0 .. ROWS*HH-1
  int row = t >> 2, hh = t & 3;
  const float4* p = (const float4*)(Wh + (size_t)row * OUT_DIM + hh * DD);
  const float4* as = (const float4*)a_src;
  const float4* ad = (const float4*)a_dst;
  float s = 0.f, d = 0.f;
#pragma unroll
  for (int i = 0; i < 16; ++i) {
    float4 v = p[i], u = as[i], w = ad[i];
    s += v.x * u.x + v.y * u.y + v.z * u.z + v.w * u.w;
    d += v.x * w.x + v.y * w.y + v.z * w.z + v.w * w.w;
  }
  e_src[row * HH + hh] = s;
  e_dst[row * HH + hh] = d;
}

// ---------------- Kernel 3: softmax attention + WMMA aggregation --------
__device__ __forceinline__ float gat_logit(float s, float d, float a) {
  float e = s + d;
  e = (e > 0.f) ? e : 0.2f * e; // leaky relu 0.2
  e += 0.5f * a;
  if (fabsf(a) < 1e-6f) e = -1e9f; // mask
  return e;
}

__global__ __launch_bounds__(128) void k_attn(const float* __restrict__ adj,
                                              const float* __restrict__ Wh,
                                              const float* __restrict__ e_src,
                                              const float* __restrict__ e_dst,
                                              float* __restrict__ attn) {
  __shared__ __align__(16) float adjc[16][260]; // 16B-aligned rows
  __shared__ __align__(16) float edstc[256][4]; // memory order [n][h]
  __shared__ float esrcs[4][16];
  __shared__ float rmaxs[4][16];
  __shared__ float rinvs[4][16];

  const int tid = threadIdx.x;
  const int b = blockIdx.x >> 6;
  const int m0 = (blockIdx.x & 63) << 4;
  const int lane = tid & 31;
  const int hh = tid >> 5; // wave == head
  const int half = lane >> 4;
  const int l = lane & 15;

  if (tid < 64) {
    int r = tid & 15, hw = tid >> 4;
    esrcs[hw][r] = e_src[(size_t)(b * NN + m0 + r) * HH + hw];
  }

  const float* adj_base = adj + ((size_t)(b * NN + m0)) * NN;
  const float* edst_base = e_dst + (size_t)(b * NN) * HH;

  // ---- phase 1: per-row online max / sum-exp (2 lanes per row) ----
  float mx = -3.0e38f, sm = 0.f;
  for (int c = 0; c < 4; ++c) {
    __syncthreads();
    const float* srcA = adj_base + c * 256;
    const float* srcD = edst_base + c * 256 * HH;
#pragma unroll
    for (int i = 0; i < 8; ++i) {
      int idx = tid + i * 128;
      int r = idx >> 6, c4 = (idx & 63) * 4;
      async_ld_b128(lds_off(&adjc[r][c4]), srcA + (size_t)r * NN + c4);
    }
#pragma unroll
    for (int i = 0; i < 2; ++i) {
      int chunk = tid + i * 128; // 256 x b128 for 1024 contiguous floats
      async_ld_b128(lds_off(&edstc[chunk][0]), srcD + chunk * 4);
    }
    wait_async0();
    __syncthreads();
    const float sv = esrcs[hh][l];
    for (int j = 0; j < 128; ++j) {
      int nl = half * 128 + j;
      float e = gat_logit(sv, edstc[nl][hh], adjc[l][nl]);
      if (e > mx) { sm *= __expf(mx - e); mx = e; }
      sm += __expf(e - mx);
    }
  }
  // combine the two half-lanes of each row (wave32 shuffle)
  float omx = __shfl_xor(mx, 16);
  float osm = __shfl_xor(sm, 16);
  float M = fmaxf(mx, omx);
  float S = sm * __expf(mx - M) + osm * __expf(omx - M);
  if (half == 0) {
    rmaxs[hh][l] = M;
    rinvs[hh][l] = 1.0f / S;
  }
  __syncthreads();

  // ---- phase 2: out[m,d] = sum_n alpha[m,n] * Wh[n,d] via fp32 WMMA ----
  v8f acc[4];
#pragma unroll
  for (int dt = 0; dt < 4; ++dt) acc[dt] = v8zero();
  const float Mv = rmaxs[hh][l];
  const float Rv = rinvs[hh][l];
  const float sv = esrcs[hh][l];

  for (int c = 0; c < 4; ++c) {
    __syncthreads();
    const float* srcA = adj_base + c * 256;
    const float* srcD = edst_base + c * 256 * HH;
#pragma unroll
    for (int i = 0; i < 8; ++i) {
      int idx = tid + i * 128;
      int r = idx >> 6, c4 = (idx & 63) * 4;
      async_ld_b128(lds_off(&adjc[r][c4]), srcA + (size_t)r * NN + c4);
    }
#pragma unroll
    for (int i = 0; i < 2; ++i) {
      int chunk = tid + i * 128;
      async_ld_b128(lds_off(&edstc[chunk][0]), srcD + chunk * 4);
    }
    wait_async0();
    __syncthreads();

    const float* whp = Wh + (size_t)(b * NN + c * 256) * OUT_DIM + hh * DD;
    for (int ks = 0; ks < 256; ks += 4) {
      const int ka = ks + 2 * half;
      float e0 = gat_logit(sv, edstc[ka][hh], adjc[l][ka]);
      float e1 = gat_logit(sv, edstc[ka + 1][hh], adjc[l][ka + 1]);
      v2f av;
      av.x = __expf(e0 - Mv) * Rv;
      av.y = __expf(e1 - Mv) * Rv;
#pragma unroll
      for (int dt = 0; dt < 4; ++dt) {
        const float* wp = whp + (size_t)ka * OUT_DIM + dt * 16 + l;
        v2f bf;
        bf.x = wp[0];
        bf.y = wp[OUT_DIM];
        acc[dt] = wmma_f32(av, bf, acc[dt]);
      }
    }
  }

  const size_t obase = (size_t)(b * NN + m0) * OUT_DIM;
#pragma unroll
  for (int dt = 0; dt < 4; ++dt)
#pragma unroll
    for (int j = 0; j < 8; ++j)
      attn[obase + (size_t)(j + half * 8) * OUT_DIM + hh * DD + dt * 16 + l] =
          acc[dt][j];
}

// ---------------- Kernel 4: BN batch stats (sum, sumsq per channel) -----
__global__ __launch_bounds__(256) void k_bnstats(const float* __restrict__ attn,
                                                 float* __restrict__ bnsum,
                                                 float* __restrict__ bnsq) {
  const int ch = threadIdx.x;
  const int r0 = blockIdx.x * 256;
  float s = 0.f, q = 0.f;
  for (int i = 0; i < 256; ++i) {
    float v = attn[(size_t)(r0 + i) * OUT_DIM + ch];
    s += v;
    q += v * v;
  }
  atomicAdd(&bnsum[ch], s);
  atomicAdd(&bnsq[ch], q);
}

// ---------------- Kernel 5: BN normalize + ELU + residual ---------------
__global__ __launch_bounds__(256) void k_final(
    const float* __restrict__ attn, const float* __restrict__ Wh,
    const float* __restrict__ bnsum, const float* __restrict__ bnsq,
    const float* __restrict__ gamma, const float* __restrict__ beta,
    float* __restrict__ out) {
  const int idx = blockIdx.x * 256 + threadIdx.x;
  const int ch = idx & 255;
  const float invn = 1.0f / (float)ROWS;
  const float mean = bnsum[ch] * invn;
  const float var = bnsq[ch] * invn - mean * mean;
  const float inv = rsqrtf(var + 1e-5f);
  const float bn = (attn[idx] - mean) * inv * gamma[ch] + beta[ch];
  const float e = (bn > 0.f) ? bn : (__expf(bn) - 1.0f);
  out[idx] = e + Wh[idx];
}

extern "C" void kernel_launch(void* const* d_in, const int* in_sizes, int n_in,
                              void* d_out, int out_size, void* d_ws,
                              size_t ws_size, hipStream_t stream) {
  (void)in_sizes; (void)n_in; (void)out_size; (void)ws_size;
  const float* h = (const float*)d_in[0];
  const float* adj = (const float*)d_in[1];
  const float* W = (const float*)d_in[2];
  const float* a_src = (const float*)d_in[3];
  const float* a_dst = (const float*)d_in[4];
  const float* gamma = (const float*)d_in[5];
  const float* beta = (const float*)d_in[6];
  float* out = (float*)d_out;

  float* ws = (float*)d_ws;
  float* Wh = ws;                                // ROWS*OUT_DIM
  float* e_src = Wh + (size_t)ROWS * OUT_DIM;    // ROWS*HH
  float* e_dst = e_src + (size_t)ROWS * HH;      // ROWS*HH
  float* attn = e_dst + (size_t)ROWS * HH;       // ROWS*OUT_DIM
  float* bnsum = attn + (size_t)ROWS * OUT_DIM;  // OUT_DIM
  float* bnsq = bnsum + OUT_DIM;                 // OUT_DIM

  hipMemsetAsync(bnsum, 0, 2 * OUT_DIM * sizeof(float), stream);

  k_wh<<<ROWS / 16, 128, 0, stream>>>(h, W, Wh);
  k_attcoef<<<(ROWS * HH) / 256, 256, 0, stream>>>(Wh, a_src, a_dst, e_src,
                                                   e_dst);
  k_attn<<<BB * (NN / 16), 128, 0, stream>>>(adj, Wh, e_src, e_dst, attn);
  k_bnstats<<<ROWS / 256, 256, 0, stream>>>(attn, bnsum, bnsq);
  k_final<<<(ROWS * OUT_DIM) / 256, 256, 0, stream>>>(attn, Wh, bnsum, bnsq,
                                                      gamma, beta, out);
}